// DigitCapsules_76828374991673
// MI455X (gfx1250) — compile-verified
//
#include <hip/hip_runtime.h>

typedef float v2f __attribute__((ext_vector_type(2)));
typedef float v4f __attribute__((ext_vector_type(4)));
typedef float v8f __attribute__((ext_vector_type(8)));

#define IN_CAPS 2048
#define OUT_CAPS 64
#define OUT_DIM 32
#define OD 2048          // OUT_CAPS * OUT_DIM
#define B_TOT 32
#define IN_DIM 16
#define GROUP_B 8        // batches per L2-resident group (8 * 16MB = 128MB < 192MB L2)
#define NGROUP 4
#define NCHUNK 128       // i-chunks per batch for deterministic partial reduction
#define CHUNK_I 16       // IN_CAPS / NCHUNK
#define ITERS 5

// ---------------------------------------------------------------------------
// GEMM: u_hat[b][i][od] = sum_k x[b][i][k] * W[i][od][k]
// WMMA mapping: M = batch (2 tiles of 16), N = od (128 tiles of 16), K = 16 (4 x K=4)
// A (16x4 f32): lane L -> row M = L%16, VGPR0 = K=(L<16?0:2), VGPR1 = K+1
// B (4x16 f32): lane L -> col N = L%16, VGPR0 = K=(L<16?0:2), VGPR1 = K+1
// D (16x16 f32): lane L -> N = L%16, VGPR r -> M = r + (L<16?0:8)  => od across
// lanes => coalesced b32 stores (each covers two full 64B lines).
// ---------------------------------------------------------------------------
__global__ __launch_bounds__(256) void caps_gemm(const float* __restrict__ x,
                                                 const float* __restrict__ W,
                                                 float* __restrict__ uhat) {
  const int i    = blockIdx.x;           // in_cap index
  const int wave = threadIdx.x >> 5;     // 0..7
  const int lane = threadIdx.x & 31;
  const int m    = lane & 15;
  const int kh   = (lane < 16) ? 0 : 2;

  // Preload A (x) for both batch tiles, all 4 K-blocks (x is tiny, L2-resident)
  v2f a[2][4];
#pragma unroll
  for (int bt = 0; bt < 2; ++bt) {
    const float* xp = x + ((size_t)(bt * 16 + m) * IN_CAPS + i) * IN_DIM + kh;
#pragma unroll
    for (int kb = 0; kb < 4; ++kb) a[bt][kb] = *(const v2f*)(xp + kb * 4);
  }

  // Each wave owns 16 disjoint od-tiles => W streamed exactly once from HBM
  for (int t = 0; t < 16; ++t) {
    const int odt = wave * 16 + t;
    const float* wp = W + ((size_t)i * OD + odt * 16 + m) * IN_DIM + kh;
    v2f bm[4];
#pragma unroll
    for (int kb = 0; kb < 4; ++kb) bm[kb] = *(const v2f*)(wp + kb * 4);

    v8f c0 = {};
    v8f c1 = {};
#pragma unroll
    for (int kb = 0; kb < 4; ++kb) {
      c0 = __builtin_amdgcn_wmma_f32_16x16x4_f32(false, a[0][kb], false, bm[kb],
                                                 (short)0, c0, false, false);
      c1 = __builtin_amdgcn_wmma_f32_16x16x4_f32(false, a[1][kb], false, bm[kb],
                                                 (short)0, c1, false, false);
    }

    const int n  = odt * 16 + m;                 // od, contiguous across lanes
    const int bh = (lane < 16) ? 0 : 8;
#pragma unroll
    for (int r = 0; r < 8; ++r) {
      uhat[((size_t)(r + bh) * IN_CAPS + i) * OD + n]      = c0[r];
      uhat[((size_t)(16 + r + bh) * IN_CAPS + i) * OD + n] = c1[r];
    }
  }
}

// ---------------------------------------------------------------------------
// Routing iteration 0: V == 0 -> softmax is exactly uniform (c = 1/64), so
// s_partial[od] = (1/64) * sum_{i in chunk} u[od].  No exp, no shuffles, low
// register pressure -> max occupancy for the one pass that streams from HBM.
// ---------------------------------------------------------------------------
__global__ __launch_bounds__(256) void caps_route_uniform(
    const float* __restrict__ uhat, float* __restrict__ part, int g) {
  const int wave = threadIdx.x >> 5;
  const int lane = threadIdx.x & 31;
  const int gw   = blockIdx.x * 8 + wave;   // 0 .. GROUP_B*NCHUNK-1
  const int bl   = gw >> 7;                 // batch within group
  const int ch   = gw & (NCHUNK - 1);       // i-chunk
  const int b    = g * GROUP_B + bl;

  const v4f zero = {0.f, 0.f, 0.f, 0.f};
  v4f sacc[16];
#pragma unroll
  for (int j = 0; j < 16; ++j) sacc[j] = zero;

  for (int ii = 0; ii < CHUNK_I; ++ii) {
    const int i = ch * CHUNK_I + ii;
    const float* up = uhat + ((size_t)b * IN_CAPS + i) * OD + lane * 4;
#pragma unroll
    for (int j = 0; j < 16; ++j) sacc[j] += *(const v4f*)(up + j * 128);
  }

  const float c = 1.0f / (float)OUT_CAPS;
  float* pp = part + ((size_t)bl * NCHUNK + ch) * OD + lane * 4;
#pragma unroll
  for (int j = 0; j < 16; ++j) *(v4f*)(pp + j * 128) = sacc[j] * c;
}

// ---------------------------------------------------------------------------
// Routing partial pass (iterations 1..4): each wave handles (batch, i-chunk).
// Lane L holds od = j*128 + L*4 + e  (j=0..15, e=0..3)  => o = j*4 + (L>>3),
// d = (L&7)*4 + e.  Softmax over o for fixed (i,d) = 16 in-lane values x the
// 4-lane group {L, L^8, L^16, L^24}  -> shfl_xor masks 8 and 16.
// Logits never materialized: b = u_hat * V  (since b_0 = 0, b_r = u_hat * V_r).
// ---------------------------------------------------------------------------
__global__ __launch_bounds__(256) void caps_route_partial(
    const float* __restrict__ uhat, const float* __restrict__ V,
    float* __restrict__ part, int g) {
  const int wave = threadIdx.x >> 5;
  const int lane = threadIdx.x & 31;
  const int gw   = blockIdx.x * 8 + wave;   // 0 .. GROUP_B*NCHUNK-1
  const int bl   = gw >> 7;                 // batch within group
  const int ch   = gw & (NCHUNK - 1);       // i-chunk
  const int b    = g * GROUP_B + bl;

  const v4f zero = {0.f, 0.f, 0.f, 0.f};
  v4f Vv[16], sacc[16];
  const float* vp = V + (size_t)b * OD + lane * 4;
#pragma unroll
  for (int j = 0; j < 16; ++j) {
    Vv[j]   = *(const v4f*)(vp + j * 128);
    sacc[j] = zero;
  }

  for (int ii = 0; ii < CHUNK_I; ++ii) {
    const int i = ch * CHUNK_I + ii;
    const float* up = uhat + ((size_t)b * IN_CAPS + i) * OD + lane * 4;
    v4f u[16];
#pragma unroll
    for (int j = 0; j < 16; ++j) u[j] = *(const v4f*)(up + j * 128);  // b128, 512B/instr

    // pass 1: per-d max of t = u * V over o
    v4f mx = {-3.4e38f, -3.4e38f, -3.4e38f, -3.4e38f};
#pragma unroll
    for (int j = 0; j < 16; ++j) {
      v4f t = u[j] * Vv[j];
      mx.x = fmaxf(mx.x, t.x); mx.y = fmaxf(mx.y, t.y);
      mx.z = fmaxf(mx.z, t.z); mx.w = fmaxf(mx.w, t.w);
    }
    mx.x = fmaxf(mx.x, __shfl_xor(mx.x, 8));  mx.x = fmaxf(mx.x, __shfl_xor(mx.x, 16));
    mx.y = fmaxf(mx.y, __shfl_xor(mx.y, 8));  mx.y = fmaxf(mx.y, __shfl_xor(mx.y, 16));
    mx.z = fmaxf(mx.z, __shfl_xor(mx.z, 8));  mx.z = fmaxf(mx.z, __shfl_xor(mx.z, 16));
    mx.w = fmaxf(mx.w, __shfl_xor(mx.w, 8));  mx.w = fmaxf(mx.w, __shfl_xor(mx.w, 16));

    // pass 2: e = exp(t-m); Z += e; overwrite u <- e*u (saves a register array)
    v4f Z = zero;
#pragma unroll
    for (int j = 0; j < 16; ++j) {
      v4f t = u[j] * Vv[j];
      v4f e;
      e.x = __expf(t.x - mx.x); e.y = __expf(t.y - mx.y);
      e.z = __expf(t.z - mx.z); e.w = __expf(t.w - mx.w);
      Z += e;
      u[j] *= e;
    }
    Z.x += __shfl_xor(Z.x, 8); Z.x += __shfl_xor(Z.x, 16);
    Z.y += __shfl_xor(Z.y, 8); Z.y += __shfl_xor(Z.y, 16);
    Z.z += __shfl_xor(Z.z, 8); Z.z += __shfl_xor(Z.z, 16);
    Z.w += __shfl_xor(Z.w, 8); Z.w += __shfl_xor(Z.w, 16);
    v4f inv;
    inv.x = 1.0f / Z.x; inv.y = 1.0f / Z.y; inv.z = 1.0f / Z.z; inv.w = 1.0f / Z.w;

#pragma unroll
    for (int j = 0; j < 16; ++j) sacc[j] += u[j] * inv;
  }

  float* pp = part + ((size_t)bl * NCHUNK + ch) * OD + lane * 4;
#pragma unroll
  for (int j = 0; j < 16; ++j) *(v4f*)(pp + j * 128) = sacc[j];
}

// ---------------------------------------------------------------------------
// Reduce partials -> s[b,o,d]; squash over d (32 lanes); update V or emit v.
// ---------------------------------------------------------------------------
__global__ __launch_bounds__(256) void caps_route_reduce(
    const float* __restrict__ part, float* __restrict__ V,
    float* __restrict__ out, int g, int last) {
  const int bl   = blockIdx.x;          // batch within group
  const int b    = g * GROUP_B + bl;
  const int wave = threadIdx.x >> 5;
  const int d    = threadIdx.x & 31;

  for (int oo = 0; oo < 8; ++oo) {
    const int o  = wave * 8 + oo;
    const int od = o * 32 + d;
    float s = 0.f;
    const float* pp = part + (size_t)bl * NCHUNK * OD + od;
#pragma unroll 8
    for (int c = 0; c < NCHUNK; ++c) s += pp[(size_t)c * OD];

    float n2 = s * s;
    n2 += __shfl_xor(n2, 1);
    n2 += __shfl_xor(n2, 2);
    n2 += __shfl_xor(n2, 4);
    n2 += __shfl_xor(n2, 8);
    n2 += __shfl_xor(n2, 16);

    // v = (n2/(1+n2)) * s / sqrt(n2 + eps)
    const float scale = n2 / ((1.0f + n2) * sqrtf(n2 + 1e-8f));
    const float v = s * scale;
    if (last) out[(size_t)b * OD + od] = v;
    else      V[(size_t)b * OD + od] += v;
  }
}

__global__ void caps_zero(float* __restrict__ p, int n) {
  const int idx = blockIdx.x * blockDim.x + threadIdx.x;
  if (idx < n) p[idx] = 0.f;
}

// ---------------------------------------------------------------------------
extern "C" void kernel_launch(void* const* d_in, const int* in_sizes, int n_in,
                              void* d_out, int out_size, void* d_ws, size_t ws_size,
                              hipStream_t stream) {
  (void)in_sizes; (void)n_in; (void)out_size; (void)ws_size;
  const float* x = (const float*)d_in[0];   // [32, 2048, 16]
  const float* W = (const float*)d_in[1];   // [2048, 64, 32, 16]
  float* out = (float*)d_out;               // [32, 64, 32]

  float* ws   = (float*)d_ws;
  float* uhat = ws;                                        // 32*2048*2048 f32 (512MB)
  float* V    = uhat + (size_t)B_TOT * IN_CAPS * OD;       // 32*2048 f32
  float* part = V + (size_t)B_TOT * OD;                    // 8*128*2048 f32 (8MB)

  caps_zero<<<(B_TOT * OD + 255) / 256, 256, 0, stream>>>(V, B_TOT * OD);
  caps_gemm<<<IN_CAPS, 256, 0, stream>>>(x, W, uhat);

  // Per L2-resident batch group, run all 5 routing iterations back-to-back so
  // iterations 2..5 re-read u_hat from L2 instead of HBM.
  const int pblocks = GROUP_B * NCHUNK / 8;
  for (int g = 0; g < NGROUP; ++g) {
    for (int r = 0; r < ITERS; ++r) {
      if (r == 0)
        caps_route_uniform<<<pblocks, 256, 0, stream>>>(uhat, part, g);
      else
        caps_route_partial<<<pblocks, 256, 0, stream>>>(uhat, V, part, g);
      caps_route_reduce<<<GROUP_B, 256, 0, stream>>>(part, V, out, g,
                                                     (r == ITERS - 1) ? 1 : 0);
    }
  }
}